// VectorQuantizer_7447473291884
// MI455X (gfx1250) — compile-verified
//
#include <hip/hip_runtime.h>
#include <hip/hip_bf16.h>

// ---------------------------------------------------------------------------
// VQ-VAE vector quantizer for MI455X (gfx1250, wave32, WMMA).
//   latents  [8,4096,128] f32   codebook [4096,128] f32
//   ema_cluster_size [4096] f32 ema_w    [4096,128] f32
// Outputs (flat, concatenated): q_ste[N*D], loss[1], indices[N](as f32),
//   new_weight[K*D], new_cluster_size[K], new_ema_w[K*D]
// ---------------------------------------------------------------------------

#define NROWS   32768          // 8*4096 latent vectors
#define DIM     128
#define KCODES  4096

typedef float v2f __attribute__((ext_vector_type(2)));
typedef float v8f __attribute__((ext_vector_type(8)));

// ---------------------------------------------------------------------------
// Kernel 0: codebook row squared norms  cn[k] = ||codebook[k]||^2
// ---------------------------------------------------------------------------
__global__ __launch_bounds__(128) void vq_cnorm_kernel(
    const float* __restrict__ cb, float* __restrict__ cn) {
  const int k = blockIdx.x;
  const int t = threadIdx.x;
  float v = cb[(size_t)k * DIM + t];
  float s = v * v;
  #pragma unroll
  for (int off = 16; off > 0; off >>= 1) s += __shfl_xor(s, off, 32);
  __shared__ float red[4];
  if ((t & 31) == 0) red[t >> 5] = s;
  __syncthreads();
  if (t == 0) cn[k] = red[0] + red[1] + red[2] + red[3];
}

// ---------------------------------------------------------------------------
// Kernel 1: fused fp32-WMMA distance GEMM + running argmin.
//   Block = 128 threads = 4 waves; block owns 64 latent rows (16 per wave).
//   A (M=16 codes x K=4 dims) from LDS-staged codebook chunk,
//   B (K=4 dims x N=16 latents) register-resident.
//   C layout: lane L, vgpr v -> (M = v + 8*(L>=16), N = L%16)  => per lane:
//   8 code-scores for ONE latent row -> in-lane argmin, one shfl_xor(16) merge.
// ---------------------------------------------------------------------------
__global__ __launch_bounds__(128) void vq_argmin_kernel(
    const float* __restrict__ latents, const float* __restrict__ codebook,
    const float* __restrict__ cnorm, int* __restrict__ idx_out) {
  __shared__ float lds_code[16 * DIM];   // 8KB codebook chunk
  __shared__ float lds_cn[16];

  const int tid  = threadIdx.x;
  const int wave = tid >> 5;
  const int lane = tid & 31;
  const int lo   = lane & 15;            // N (latent) / M (code row) selector
  const int hi   = lane >> 4;            // K-dim pair selector within fragments
  const int rowbase = blockIdx.x * 64 + wave * 16;

  // Preload B fragments: latent strip rows, register resident (64 VGPRs).
  v2f bfrag[32];
  const float* lrow = latents + (size_t)(rowbase + lo) * DIM;
  #pragma unroll
  for (int j = 0; j < 32; ++j) {
    bfrag[j] = *(const v2f*)(lrow + 4 * j + 2 * hi);
  }

  float best_val = 3.0e38f;
  int   best_idx = 0;

  for (int kb = 0; kb < KCODES; kb += 16) {
    __syncthreads();  // protect LDS from previous iteration's readers
    // Cooperative stage of 16 codes x 128 dims (512 float4, 4 per thread).
    const float4* src = (const float4*)(codebook + (size_t)kb * DIM);
    float4* dst = (float4*)lds_code;
    #pragma unroll
    for (int t = 0; t < 4; ++t) dst[tid + 128 * t] = src[tid + 128 * t];
    if (tid < 16) lds_cn[tid] = cnorm[kb + tid];
    __syncthreads();

    v8f acc = {};
    #pragma unroll
    for (int j = 0; j < 32; ++j) {       // D=128 in K=4 steps
      v2f a = *(const v2f*)(lds_code + lo * DIM + 4 * j + 2 * hi);
      acc = __builtin_amdgcn_wmma_f32_16x16x4_f32(
          /*neg_a=*/false, a, /*neg_b=*/false, bfrag[j],
          /*c_mod=*/(short)0, acc, /*reuse_a=*/false, /*reuse_b=*/false);
    }

    // score = ||c||^2 - 2 * dot   (row norm is argmin-invariant, dropped)
    #pragma unroll
    for (int v = 0; v < 8; ++v) {
      const int m = v + 8 * hi;
      const float score = lds_cn[m] - 2.0f * acc[v];
      const int ci = kb + m;
      if (score < best_val) { best_val = score; best_idx = ci; }
    }
  }

  // Merge the two half-wave code ranges (lane L holds codes m<8, L^16 m>=8).
  const float oval = __shfl_xor(best_val, 16, 32);
  const int   oidx = __shfl_xor(best_idx, 16, 32);
  if (oval < best_val || (oval == best_val && oidx < best_idx)) {
    best_val = oval; best_idx = oidx;
  }
  if (hi == 0) idx_out[rowbase + lo] = best_idx;
}

// ---------------------------------------------------------------------------
// Kernel 2: gather quantized rows, write q_ste (== quantized numerically),
// accumulate loss sum, scatter segment sums (counts, dw), emit indices.
// ---------------------------------------------------------------------------
__global__ __launch_bounds__(128) void vq_gather_kernel(
    const float* __restrict__ latents, const float* __restrict__ codebook,
    const int* __restrict__ idx, float* __restrict__ q_out,
    float* __restrict__ idx_out_f, float* __restrict__ dw,
    float* __restrict__ counts, float* __restrict__ loss_sum) {
  const int n = blockIdx.x;
  const int d = threadIdx.x;
  const int k = idx[n];
  const float x = latents[(size_t)n * DIM + d];
  const float q = codebook[(size_t)k * DIM + d];
  q_out[(size_t)n * DIM + d] = q;        // x + stopgrad(q - x) == q
  const float diff = x - q;
  float s = diff * diff;
  #pragma unroll
  for (int off = 16; off > 0; off >>= 1) s += __shfl_xor(s, off, 32);
  __shared__ float red[4];
  if ((d & 31) == 0) red[d >> 5] = s;
  __syncthreads();
  if (d == 0) {
    atomicAdd(loss_sum, red[0] + red[1] + red[2] + red[3]);
    atomicAdd(&counts[k], 1.0f);
    idx_out_f[n] = (float)k;
  }
  atomicAdd(&dw[(size_t)k * DIM + d], x);   // segment_sum(flat, idx)
}

// ---------------------------------------------------------------------------
// Kernel 3: new_cluster_size + global sum n.
// ---------------------------------------------------------------------------
__global__ __launch_bounds__(256) void vq_cs_kernel(
    const float* __restrict__ ema_cs, const float* __restrict__ counts,
    float* __restrict__ ncs_out, float* __restrict__ n_sum) {
  const int k = blockIdx.x * blockDim.x + threadIdx.x;
  if (k >= KCODES) return;
  const float v = 0.99f * ema_cs[k] + 0.01f * counts[k];
  ncs_out[k] = v;
  float s = v;
  #pragma unroll
  for (int off = 16; off > 0; off >>= 1) s += __shfl_xor(s, off, 32);
  if ((threadIdx.x & 31) == 0) atomicAdd(n_sum, s);
}

// ---------------------------------------------------------------------------
// Kernel 4: new_ema_w, normalized new_weight, and the scalar loss.
// ---------------------------------------------------------------------------
__global__ __launch_bounds__(256) void vq_update_kernel(
    const float* __restrict__ ema_w, const float* __restrict__ dw,
    const float* __restrict__ ncs, const float* __restrict__ n_sum,
    const float* __restrict__ loss_sum, float* __restrict__ new_ema_w,
    float* __restrict__ new_weight, float* __restrict__ loss_out) {
  const size_t i = (size_t)blockIdx.x * blockDim.x + threadIdx.x;  // K*D
  const int k = (int)(i >> 7);
  const float nw = 0.99f * ema_w[i] + 0.01f * dw[i];
  new_ema_w[i] = nw;
  const float n = *n_sum;
  const float cs = (ncs[k] + 1e-5f) / (n + (float)KCODES * 1e-5f) * n;
  new_weight[i] = nw / cs;
  if (i == 0) {
    // commitment(0.25) + codebook(1.0) both are mean((x-q)^2)
    *loss_out = 1.25f * (*loss_sum) / (float)((size_t)NROWS * DIM);
  }
}

// ---------------------------------------------------------------------------
extern "C" void kernel_launch(void* const* d_in, const int* in_sizes, int n_in,
                              void* d_out, int out_size, void* d_ws,
                              size_t ws_size, hipStream_t stream) {
  const float* latents  = (const float*)d_in[0];
  const float* codebook = (const float*)d_in[1];
  const float* ema_cs   = (const float*)d_in[2];
  const float* ema_w    = (const float*)d_in[3];

  float* out = (float*)d_out;
  float* q_out      = out;                            // N*D = 4194304
  float* loss_out   = q_out + (size_t)NROWS * DIM;    // 1
  float* idx_out_f  = loss_out + 1;                   // N = 32768
  float* weight_out = idx_out_f + NROWS;              // K*D = 524288
  float* ncs_out    = weight_out + (size_t)KCODES * DIM;  // K = 4096
  float* nemaw_out  = ncs_out + KCODES;               // K*D = 524288

  char* ws = (char*)d_ws;
  int*   idx_ws   = (int*)ws;     ws += (size_t)NROWS * sizeof(int);
  float* cnorm_ws = (float*)ws;   ws += (size_t)KCODES * sizeof(float);
  float* counts_ws = (float*)ws;  ws += (size_t)KCODES * sizeof(float);
  float* dw_ws     = (float*)ws;  ws += (size_t)KCODES * DIM * sizeof(float);
  float* loss_ws   = (float*)ws;  ws += sizeof(float);
  float* nsum_ws   = (float*)ws;

  // Zero the atomic accumulation region (counts, dw, loss, n) — contiguous.
  hipMemsetAsync(counts_ws, 0,
                 ((size_t)KCODES + (size_t)KCODES * DIM + 2) * sizeof(float),
                 stream);

  vq_cnorm_kernel<<<KCODES, 128, 0, stream>>>(codebook, cnorm_ws);
  vq_argmin_kernel<<<NROWS / 64, 128, 0, stream>>>(latents, codebook, cnorm_ws,
                                                   idx_ws);
  vq_gather_kernel<<<NROWS, 128, 0, stream>>>(latents, codebook, idx_ws, q_out,
                                              idx_out_f, dw_ws, counts_ws,
                                              loss_ws);
  vq_cs_kernel<<<KCODES / 256, 256, 0, stream>>>(ema_cs, counts_ws, ncs_out,
                                                 nsum_ws);
  vq_update_kernel<<<(KCODES * DIM) / 256, 256, 0, stream>>>(
      ema_w, dw_ws, ncs_out, nsum_ws, loss_ws, nemaw_out, weight_out, loss_out);
}